// UVAlignCell_1176821039486
// MI455X (gfx1250) — compile-verified
//
#include <hip/hip_runtime.h>
#include <hip/hip_bf16.h>

typedef float v2f __attribute__((ext_vector_type(2)));
typedef float v8f __attribute__((ext_vector_type(8)));

#define HXW   (192 * 192)
#define Z_N   (16 * 64 * HXW)     // 37,748,736
#define HC_N  (16 * 8 * HXW)      // 4,718,592

__device__ __forceinline__ float sigmoidf_(float x) {
    return 1.0f / (1.0f + __expf(-x));
}
__device__ __forceinline__ float tanhf_(float x) {
    // 1 - 2/(e^{2x}+1): safe at +/- inf
    float e = __expf(2.0f * x);
    return 1.0f - 2.0f / (e + 1.0f);
}

// ---------------------------------------------------------------------------
// Kernel 1: fused 1x1-conv gates (WMMA f32 16x16x4) + LSTM pointwise.
// One wave handles 16 pixels (contiguous in W) x 32 output gate channels.
//   A (16x4)  = weight tile  (M = gate channel, K = input channel)
//   B (4x16)  = comb tile    (K = input channel, N = pixel)
//   D (16x16) = gates tile
// f32 A 16x4 layout: lane<16 holds K={k0,k0+1} for M=lane; lane>=16 K={k0+2,k0+3}.
// f32 D 16x16 layout: VGPR r -> M=r (lanes 0-15) / M=r+8 (lanes 16-31), N=lane%16.
// ---------------------------------------------------------------------------
__global__ void k_gates_lstm(const float* __restrict__ x,
                             const float* __restrict__ hid,
                             const float* __restrict__ cell,
                             const float* __restrict__ wr,   // (32,72)
                             const float* __restrict__ br,   // (32)
                             float* __restrict__ hidden_out,
                             float* __restrict__ cell_out) {
    const int lane = threadIdx.x & 31;
    const int l15  = lane & 15;
    const int hi   = lane >> 4;                 // 0: lanes 0-15, 1: lanes 16-31
    const int wave = threadIdx.x >> 5;

    const int t  = blockIdx.x * 8 + wave;       // 36864 pixel tiles total
    const int wt = t % 12;
    const int hh = (t / 12) % 192;
    const int b  = t / (12 * 192);
    const int pixoff = hh * 192 + wt * 16 + l15;

    const float* w0 = wr + l15 * 72;            // M-tile 0: channels 0..15
    const float* w1 = wr + (l15 + 16) * 72;     // M-tile 1: channels 16..31

    v8f acc0, acc1;                             // bias preload into C
#pragma unroll
    for (int r = 0; r < 8; ++r) {
        acc0[r] = br[r + (hi ? 8 : 0)];
        acc1[r] = br[16 + r + (hi ? 8 : 0)];
    }

    const float* xb = x   + (size_t)b * 64 * HXW + pixoff;
    const float* hb = hid + (size_t)b * 8 * HXW + pixoff;

#pragma unroll
    for (int kb = 0; kb < 18; ++kb) {           // K = 72 = 18 * 4
        const int k0 = kb * 4;
        const int ka = k0 + 2 * hi;             // this lane-half's K pair
        float bx, by;
        if (ka < 64) {                          // uniform per kb
            bx = xb[(size_t)ka * HXW];
            by = xb[(size_t)(ka + 1) * HXW];
        } else {
            bx = hb[(size_t)(ka - 64) * HXW];
            by = hb[(size_t)(ka - 63) * HXW];
        }
        v2f bv = {bx, by};
        v2f a0 = {w0[ka], w0[ka + 1]};
        v2f a1 = {w1[ka], w1[ka + 1]};
        acc0 = __builtin_amdgcn_wmma_f32_16x16x4_f32(false, a0, false, bv,
                                                     (short)0, acc0, false, false);
        acc1 = __builtin_amdgcn_wmma_f32_16x16x4_f32(false, a1, false, bv,
                                                     (short)0, acc1, false, false);
    }

    // Gate split by channel: i=0..7, f=8..15, g=16..23, o=24..31.
    // acc0[r]: lanes<16 = i_r, lanes>=16 = f_r ; acc1[r]: g_r / o_r.
    const float* cb = cell + (size_t)b * 8 * HXW + pixoff;
#pragma unroll
    for (int r = 0; r < 8; ++r) {
        float v0 = acc0[r];
        float v1 = acc1[r];
        float s0 = __shfl_xor(v0, 16, 32);
        float s1 = __shfl_xor(v1, 16, 32);
        float ig = hi ? s0 : v0;
        float fg = hi ? v0 : s0;
        float gg = hi ? s1 : v1;
        float og = hi ? v1 : s1;
        float cold = cb[(size_t)r * HXW];
        float cn = sigmoidf_(fg) * cold + sigmoidf_(ig) * tanhf_(gg);
        float hn = sigmoidf_(og) * tanhf_(cn);
        if (!hi) {
            size_t off = (size_t)(b * 8 + r) * HXW + pixoff;
            hidden_out[off] = hn;
            cell_out[off]   = cn;
        }
    }
}

// ---------------------------------------------------------------------------
// Kernel 2: 7x7 VALID conv (2 out ch) + sigmoid + fused u/v partial sums.
// 16x16 output tile per block; 22x22x8 input tile + weights in LDS.
// ---------------------------------------------------------------------------
__global__ void k_conv_reduce(const float* __restrict__ hn,
                              const float* __restrict__ wc,   // (2,8,7,7)
                              const float* __restrict__ bc,   // (2)
                              float* __restrict__ usum,       // (16,3)
                              float* __restrict__ vsum) {     // (16,3)
    __shared__ float tile[8][22][22];
    __shared__ float wsm[784];
    __shared__ float su[3], sv[3];

    const int tid = threadIdx.x;
    const int bx  = blockIdx.x;                 // 16 * 12 * 12 blocks
    const int tw  = bx % 12;
    const int th  = (bx / 12) % 12;
    const int b   = bx / 144;
    const int h0  = th * 16, w0 = tw * 16;

    for (int i = tid; i < 784; i += 256) wsm[i] = wc[i];
    for (int i = tid; i < 8 * 22 * 22; i += 256) {
        int ic = i / 484, rem = i % 484;
        int r = rem / 22, c = rem % 22;
        int gh = h0 + r, gw = w0 + c;
        float v = 0.0f;
        if (gh < 192 && gw < 192)
            v = hn[((size_t)(b * 8 + ic) * 192 + gh) * 192 + gw];
        (&tile[0][0][0])[i] = v;
    }
    if (tid < 3) { su[tid] = 0.0f; sv[tid] = 0.0f; }
    __syncthreads();

    const int ty = tid >> 4, tx = tid & 15;
    const int oh = h0 + ty, ow = w0 + tx;
    if (oh < 186 && ow < 186) {
        float a0 = bc[0], a1 = bc[1];
        for (int ic = 0; ic < 8; ++ic) {
            const float* wp0 = &wsm[ic * 49];
            const float* wp1 = &wsm[392 + ic * 49];
#pragma unroll
            for (int ky = 0; ky < 7; ++ky) {
#pragma unroll
                for (int kx = 0; kx < 7; ++kx) {
                    float v = tile[ic][ty + ky][tx + kx];
                    a0 = __builtin_fmaf(v, wp0[ky * 7 + kx], a0);
                    a1 = __builtin_fmaf(v, wp1[ky * 7 + kx], a1);
                }
            }
        }
        atomicAdd(&su[ow / 62], sigmoidf_(a0));   // u buckets: w-thirds of ch0
        atomicAdd(&sv[oh / 62], sigmoidf_(a1));   // v buckets: h-thirds of ch1
    }
    __syncthreads();
    if (tid < 3)       atomicAdd(&usum[b * 3 + tid], su[tid]);
    else if (tid < 6)  atomicAdd(&vsum[b * 3 + (tid - 3)], sv[tid - 3]);
}

// ---------------------------------------------------------------------------
// Kernel 3a: zero the atomic accumulators in workspace.
// ---------------------------------------------------------------------------
__global__ void k_init_ws(float* __restrict__ ws) {
    ws[threadIdx.x] = 0.0f;                     // 96 floats
}

// ---------------------------------------------------------------------------
// Kernel 3b: per-batch 9 stencil coefficients coeff[b][dv][du] = v_dv * u_du.
// ---------------------------------------------------------------------------
__global__ void k_coeff(const float* __restrict__ usum,
                        const float* __restrict__ vsum,
                        float* __restrict__ coeff) {
    int t = threadIdx.x;
    if (t < 144) {
        int b = t / 9, r = t % 9, dv = r / 3, du = r % 3;
        const float invN = 1.0f / (186.0f * 62.0f);
        coeff[t] = (vsum[b * 3 + dv] * invN) * (usum[b * 3 + du] * invN);
    }
}

// ---------------------------------------------------------------------------
// Kernel 4: fused separable 3-tap (u horizontal then v vertical, zero pads)
// = 9-tap stencil with per-batch coefficients. Pure streaming.
// ---------------------------------------------------------------------------
__global__ void k_stencil(const float* __restrict__ x,
                          const float* __restrict__ coeff,
                          float* __restrict__ z) {
    size_t idx = (size_t)blockIdx.x * 256 + threadIdx.x;   // < Z_N
    int w = (int)(idx % 192);
    int h = (int)((idx / 192) % 192);
    int b = (int)(idx / ((size_t)64 * HXW));
    const float* c = coeff + b * 9;
    const float* p = x + idx;

    bool hm = h > 0, hp = h < 191, wm = w > 0, wp = w < 191;
    float acc = c[4] * p[0];
    if (wm) acc = __builtin_fmaf(c[3], p[-1], acc);
    if (wp) acc = __builtin_fmaf(c[5], p[1], acc);
    if (hm) {
        acc = __builtin_fmaf(c[1], p[-192], acc);
        if (wm) acc = __builtin_fmaf(c[0], p[-193], acc);
        if (wp) acc = __builtin_fmaf(c[2], p[-191], acc);
    }
    if (hp) {
        acc = __builtin_fmaf(c[7], p[192], acc);
        if (wm) acc = __builtin_fmaf(c[6], p[191], acc);
        if (wp) acc = __builtin_fmaf(c[8], p[193], acc);
    }
    z[idx] = acc;
}

// ---------------------------------------------------------------------------
extern "C" void kernel_launch(void* const* d_in, const int* in_sizes, int n_in,
                              void* d_out, int out_size, void* d_ws, size_t ws_size,
                              hipStream_t stream) {
    const float* x      = (const float*)d_in[0];
    const float* hidden = (const float*)d_in[1];
    const float* cell   = (const float*)d_in[2];
    const float* w_rnn  = (const float*)d_in[3];
    const float* b_rnn  = (const float*)d_in[4];
    const float* w_conv = (const float*)d_in[5];
    const float* b_conv = (const float*)d_in[6];

    float* out = (float*)d_out;
    float* z   = out;                 // (16,64,192,192)
    float* hn  = out + Z_N;           // (16,8,192,192)
    float* cn  = out + Z_N + HC_N;    // (16,8,192,192)

    float* ws    = (float*)d_ws;
    float* usum  = ws;                // 48 floats
    float* vsum  = ws + 48;           // 48 floats
    float* coeff = ws + 96;           // 144 floats

    k_init_ws   <<<1, 96, 0, stream>>>(ws);
    k_gates_lstm<<<36864 / 8, 256, 0, stream>>>(x, hidden, cell, w_rnn, b_rnn, hn, cn);
    k_conv_reduce<<<16 * 12 * 12, 256, 0, stream>>>(hn, w_conv, b_conv, usum, vsum);
    k_coeff     <<<1, 160, 0, stream>>>(usum, vsum, coeff);
    k_stencil   <<<Z_N / 256, 256, 0, stream>>>(x, coeff, z);
}